// RoIPooling2D_73693048865545
// MI455X (gfx1250) — compile-verified
//
#include <hip/hip_runtime.h>
#include <hip/hip_bf16.h>

#define RPH 7
#define RPW 7

// Problem dims are fixed by the reference setup; make them compile-time so all
// divisions fold to magic-multiply/shift and address math becomes shifts.
constexpr int kC = 256;
constexpr int kH = 64;
constexpr int kW = 64;
constexpr int kBinsPerChan = RPH * RPW;            // 49
constexpr int kElemsPerRoi = kC * kBinsPerChan;    // 12544
constexpr int kBlocksPerRoi = (kElemsPerRoi + 255) / 256;  // 49 (exact)

__device__ __forceinline__ float clampf(float v, float lo, float hi) {
    return fminf(fmaxf(v, lo), hi);
}

__global__ __launch_bounds__(256) void roipool_max_kernel(
    const float* __restrict__ feat,   // [N, 256, 64, 64]
    const float* __restrict__ rois,   // [R, 5] (bidx, x0, y0, x1, y1) image coords
    float* __restrict__ out)          // [R, 256, 7, 7]
{
    int roi  = blockIdx.x / kBlocksPerRoi;           // const divisor -> mul/shift
    int part = blockIdx.x - roi * kBlocksPerRoi;
    int flat = part * 256 + threadIdx.x;             // [0, 12544)

    int c   = flat / kBinsPerChan;                   // const divisor -> mul/shift
    int bin = flat - c * kBinsPerChan;
    int ph  = bin / RPW;
    int pw  = bin - ph * RPW;

    // ROI row is uniform across the block -> broadcast-friendly loads.
    const float* rr = rois + (size_t)roi * 5;
    int   b  = (int)rr[0];
    float x0 = clampf(floorf(rr[1] * 0.03125f), 0.0f, (float)kW);
    float y0 = clampf(floorf(rr[2] * 0.03125f), 0.0f, (float)kH);
    float x1 = clampf(floorf(rr[3] * 0.03125f), 0.0f, (float)kW);
    float y1 = clampf(floorf(rr[4] * 0.03125f), 0.0f, (float)kH);
    bool valid = (x1 > x0) && (y1 > y0);

    float bin_h = (y1 - y0) / 7.0f;   // IEEE division to match the reference exactly
    float bin_w = (x1 - x0) / 7.0f;

    // astype(int32) on non-negative values == C truncation cast.
    int ys = min(max((int)(y0 + floorf((float)ph * bin_h)), 0), kH);
    int ye = min(max((int)(y0 + ceilf(((float)ph + 1.0f) * bin_h)), 0), kH);
    int xs = min(max((int)(x0 + floorf((float)pw * bin_w)), 0), kW);
    int xe = min(max((int)(x0 + ceilf(((float)pw + 1.0f) * bin_w)), 0), kW);

    bool nonempty = valid && (ye > ys) && (xe > xs);

    float result = 0.0f;
    if (nonempty) {
        // (b*256 + c) * 64 * 64 -> pure shifts with compile-time dims.
        const float* base = feat + ((size_t)(b * kC + c)) * (kH * kW);

        // Non-blocking prefetch of every row segment we will touch
        // (global_prefetch_b8 on gfx1250), then the dependent max loops.
        for (int y = ys; y < ye; ++y) {
            __builtin_prefetch(base + y * kW + xs, 0, 0);
        }

        float m = -INFINITY;
        for (int y = ys; y < ye; ++y) {
            const float* row = base + y * kW;
            for (int x = xs; x < xe; ++x) {
                m = fmaxf(m, row[x]);
            }
        }
        result = m;   // region non-empty and inputs finite -> finite
    }

    // out layout [R, C, PH, PW]; flat == c*49 + ph*7 + pw
    out[(size_t)roi * kElemsPerRoi + flat] = result;
}

extern "C" void kernel_launch(void* const* d_in, const int* in_sizes, int n_in,
                              void* d_out, int out_size, void* d_ws, size_t ws_size,
                              hipStream_t stream) {
    const float* feat = (const float*)d_in[0];
    const float* rois = (const float*)d_in[1];
    float* out = (float*)d_out;

    int R = in_sizes[1] / 5;                       // 128

    dim3 grid((unsigned)(R * kBlocksPerRoi));      // 128 * 49 = 6272 blocks
    dim3 block(256);                                // 8 wave32 per block

    roipool_max_kernel<<<grid, block, 0, stream>>>(feat, rois, out);
}